// SpectralFeatureExtractor_90580860273233
// MI455X (gfx1250) — compile-verified
//
#include <hip/hip_runtime.h>
#include <math.h>

typedef float v2f __attribute__((ext_vector_type(2)));
typedef float v8f __attribute__((ext_vector_type(8)));

#define B_SZ   512
#define L_SZ   37500
#define NPS    1000
#define HSTEP  500
#define NT     74
#define NF     501
#define NFP    512
#define TOTF   (B_SZ*NT)     // 37888
#define KC     40
#define SA_STR 44
#define PSD_NORM (1.0f/46875.0f)   // fs * sum(win^2) = 125 * 375

// ---- d_out layout (floats) ----
#define OUT_MEL      0
#define OUT_SPECTRAL (B_SZ*64*NT)                 // 2424832
#define OUT_RFEAT    (OUT_SPECTRAL + B_SZ*32*NF)  // 10633216
#define OUT_CFEAT    (OUT_RFEAT + B_SZ*32)        // 10649600
#define OUT_SPEC     (OUT_CFEAT + B_SZ*32)        // 10665984

// ---- workspace layout (float offsets; doubles 8B-aligned) ----
#define WS_TRIGC 0
#define WS_TRIGS (NFP*NPS)           // 512000
#define WS_WIN   (2*NFP*NPS)         // 1024000
#define WS_FMEAN (WS_WIN + 1024)     // 1025024
#define WS_ENV   (WS_FMEAN + TOTF)   // 1062912
#define WS_DSUMF (WS_ENV + B_SZ*NF)  // 1319424 floats -> byte 5277696 (8B aligned)
#define WS_BNP   (WS_DSUMF + 224)    // after 112 doubles
// dsums doubles: [0:32) spec sum, [32:64) spec sumsq, [64:88) card sum, [88:112) card sumsq
// bnp floats:    [0:32) smean,    [32:64) sinv,       [64:88) cmean,    [88:112) cinv

struct MelMap { int c[64]; };

// ---------------- init: zero BN accumulators ----------------
__global__ void k_init(double* dsums) {
  int i = threadIdx.x;
  if (i < 112) dsums[i] = 0.0;
}

// ---------------- DFT matrices + Hann window ----------------
__global__ void k_trig(float* ws) {
  int i = blockIdx.x * 256 + threadIdx.x;
  if (i < NFP * NPS) {
    int k = i / NPS, n = i - k * NPS;
    int p = (int)(((long long)k * (long long)n) % NPS);   // exact reduction
    if (p > NPS / 2) p -= NPS;
    double th = 6.283185307179586 * (double)p / (double)NPS;
    ws[WS_TRIGC + i] = (float)cos(th);
    ws[WS_TRIGS + i] = (float)sin(th);
  }
  if (i < NPS) {
    double th = 6.283185307179586 * (double)i / (double)NPS;
    ws[WS_WIN + i] = (float)(0.5 - 0.5 * cos(th));
  }
}

// ---------------- per-frame mean (detrend) ----------------
__global__ void k_fmean(const float* __restrict__ x, float* __restrict__ ws) {
  __shared__ float red[256];
  int f = blockIdx.x;
  int b = f / NT, t = f - b * NT;
  const float* p = x + (size_t)b * L_SZ + (size_t)t * HSTEP;
  float s = 0.f;
  for (int i = threadIdx.x; i < NPS; i += 256) s += p[i];
  red[threadIdx.x] = s; __syncthreads();
  for (int st = 128; st > 0; st >>= 1) {
    if (threadIdx.x < st) red[threadIdx.x] += red[threadIdx.x + st];
    __syncthreads();
  }
  if (threadIdx.x == 0) ws[WS_FMEAN + f] = red[0] * (1.0f / NPS);
}

// ---------------- DFT-as-GEMM via V_WMMA_F32_16X16X4_F32 ----------------
// block tile: 64 frames (M) x 32 freqs (N); 8 waves = 4(M) x 2(N); K chunk 40 in LDS
__launch_bounds__(256)
__global__ void k_dft(const float* __restrict__ x, const float* __restrict__ ws,
                      float* __restrict__ out) {
  const float* trigC = ws + WS_TRIGC;
  const float* trigS = ws + WS_TRIGS;
  const float* win   = ws + WS_WIN;
  const float* fmean = ws + WS_FMEAN;
  float* spec = out + OUT_SPEC;

  __shared__ float sA[64 * SA_STR];        // frames [m][k], pad 44 -> conflict-free WMMA reads
  __shared__ float sB[2 * 32 * SA_STR];    // cos then sin [n][k]

  const int tid  = threadIdx.x;
  const int lane = tid & 31;
  const int wave = tid >> 5;
  const int wm = wave >> 1;                // 0..3
  const int wn = wave & 1;                 // 0..1
  const int h  = lane >> 4;                // lane-half: K pair select
  const int lm = lane & 15;

  const int M0 = blockIdx.x * 64;
  const int N0 = blockIdx.y * 32;
  const int m64 = wm * 16 + lm;
  const int n32 = wn * 16 + lm;

  v8f accR = {0.f,0.f,0.f,0.f,0.f,0.f,0.f,0.f};
  v8f accI = {0.f,0.f,0.f,0.f,0.f,0.f,0.f,0.f};

  for (int k0 = 0; k0 < NPS; k0 += KC) {
    __syncthreads();
    // stage A: 64 x 40, fused de-mean + Hann window (k fastest -> coalesced)
    for (int e = tid; e < 64 * KC; e += 256) {           // 10 iters exact
      int m = e / KC, k = e - m * KC;
      int fm = M0 + m;
      int b = fm / NT, t = fm - b * NT;
      float v = (x[(size_t)b * L_SZ + (size_t)t * HSTEP + k0 + k] - fmean[fm]) * win[k0 + k];
      sA[m * SA_STR + k] = v;
    }
    // stage B: cos & sin tiles, 2 x 32 x 40
    for (int e = tid; e < 2 * 32 * KC; e += 256) {       // 10 iters exact
      int which = e / (32 * KC);
      int r = e - which * (32 * KC);
      int n = r / KC, k = r - n * KC;
      const float* tg = which ? trigS : trigC;
      sB[which * (32 * SA_STR) + n * SA_STR + k] = tg[(size_t)(N0 + n) * NPS + k0 + k];
    }
    __syncthreads();
    #pragma unroll
    for (int s = 0; s < KC / 4; ++s) {
      int kb = s * 4 + 2 * h;                 // K base per ISA 16x16x4 layout
      v2f a, bc, bs;
      a.x  = sA[m64 * SA_STR + kb];
      a.y  = sA[m64 * SA_STR + kb + 1];
      bc.x = sB[n32 * SA_STR + kb];
      bc.y = sB[n32 * SA_STR + kb + 1];
      bs.x = sB[32 * SA_STR + n32 * SA_STR + kb];
      bs.y = sB[32 * SA_STR + n32 * SA_STR + kb + 1];
      accR = __builtin_amdgcn_wmma_f32_16x16x4_f32(false, a, false, bc, (short)0, accR, false, false);
      accI = __builtin_amdgcn_wmma_f32_16x16x4_f32(false, a, false, bs, (short)0, accI, false, false);
    }
  }

  // epilogue: psd -> log, transpose through LDS, coalesced store along t
  __syncthreads();
  float* sT = sA;                            // reuse; 32 x 68 fits in 64*44
  const int kf = N0 + n32;
  const float sc = (kf == 0 || kf == NF - 1) ? 1.0f : 2.0f;
  #pragma unroll
  for (int r = 0; r < 8; ++r) {
    int mloc = wm * 16 + 8 * h + r;          // D layout: VGPR r -> M = r + 8*half
    float R = accR[r], I = accI[r];
    float v = logf((R * R + I * I) * PSD_NORM * sc + 1e-8f);
    sT[n32 * 68 + mloc] = v;
  }
  __syncthreads();
  for (int e = tid; e < 32 * 64; e += 256) { // 8 iters exact
    int n = e >> 6, m = e & 63;
    int k = N0 + n;
    if (k < NF) {
      int fm = M0 + m;
      int b = fm / NT, t = fm - b * NT;
      spec[(size_t)b * (NF * NT) + (size_t)k * NT + t] = sT[n * 68 + m];
    }
  }
}

// ---------------- env = spec.mean(t) ----------------
__global__ void k_env(const float* __restrict__ out, float* __restrict__ ws) {
  int i = blockIdx.x * 256 + threadIdx.x;     // exactly B*NF
  int b = i / NF, k = i - b * NF;
  const float* p = out + OUT_SPEC + (size_t)b * (NF * NT) + (size_t)k * NT;
  float s = 0.f;
  for (int t = 0; t < NT; ++t) s += p[t];
  ws[WS_ENV + i] = s * (1.0f / NT);
}

// ---------------- mel: degenerate triangle filters == row gather ----------------
__global__ void k_mel(float* __restrict__ out, MelMap mm) {
  int i = blockIdx.x * 256 + threadIdx.x;     // exactly B*64*NT
  int t = i % NT;
  int m = (i / NT) & 63;
  int b = i / (NT * 64);
  int c = mm.c[m];
  out[OUT_MEL + i] = (c >= 0)
      ? out[OUT_SPEC + ((size_t)b * NF + c) * NT + t] : 0.0f;
}

// ---------------- spectral branch: pass1 BN sums (double atomics) ----------------
__global__ void k_spec_sums(const float* __restrict__ ws, const float* __restrict__ w,
                            const float* __restrict__ bias, double* __restrict__ dsums) {
  __shared__ double sacc[64];
  int tid = threadIdx.x, lane = tid & 31;
  if (tid < 64) sacc[tid] = 0.0;
  __syncthreads();
  int i = blockIdx.x * 256 + tid;             // exactly B*NF
  int b = i / NF, l = i - b * NF;
  const float* env = ws + WS_ENV;
  float e[5];
  #pragma unroll
  for (int j = 0; j < 5; ++j) {
    int idx = l + j - 2;
    e[j] = (idx >= 0 && idx < NF) ? env[b * NF + idx] : 0.0f;
  }
  #pragma unroll
  for (int c = 0; c < 32; ++c) {
    float y = bias[c];
    #pragma unroll
    for (int j = 0; j < 5; ++j) y += w[c * 5 + j] * e[j];
    float y2 = y * y;
    for (int o = 16; o > 0; o >>= 1) { y += __shfl_xor(y, o); y2 += __shfl_xor(y2, o); }
    if (lane == 0) { atomicAdd(&sacc[c], (double)y); atomicAdd(&sacc[32 + c], (double)y2); }
  }
  __syncthreads();
  if (tid < 64) atomicAdd(&dsums[tid], sacc[tid]);
}

// ---------------- cardiac branch: pass1 BN sums ----------------
__global__ void k_card_sums(const float* __restrict__ ws, const float* __restrict__ w,
                            const float* __restrict__ bias, double* __restrict__ dsums) {
  __shared__ double sacc[48];
  int tid = threadIdx.x, lane = tid & 31;
  if (tid < 48) sacc[tid] = 0.0;
  __syncthreads();
  int i = blockIdx.x * 256 + tid;             // exactly B*18
  int b = i / 18, j = i - b * 18;
  const float* env = ws + WS_ENV;
  float e[7];
  #pragma unroll
  for (int t = 0; t < 7; ++t) {
    int idx = j + t - 3;
    e[t] = (idx >= 0 && idx < 18) ? env[b * NF + 6 + idx] : 0.0f;
  }
  #pragma unroll
  for (int c = 0; c < 24; ++c) {
    float y = bias[c];
    #pragma unroll
    for (int t = 0; t < 7; ++t) y += w[c * 7 + t] * e[t];
    float y2 = y * y;
    for (int o = 16; o > 0; o >>= 1) { y += __shfl_xor(y, o); y2 += __shfl_xor(y2, o); }
    if (lane == 0) { atomicAdd(&sacc[c], (double)y); atomicAdd(&sacc[24 + c], (double)y2); }
  }
  __syncthreads();
  if (tid < 24) atomicAdd(&dsums[64 + tid], sacc[tid]);
  else if (tid < 48) atomicAdd(&dsums[88 + (tid - 24)], sacc[tid]);
}

// ---------------- BN finalize ----------------
__global__ void k_bnfin(const double* __restrict__ dsums, float* __restrict__ bnp) {
  int c = threadIdx.x;
  if (c < 32) {
    double n = (double)B_SZ * NF;
    double m = dsums[c] / n;
    double v = dsums[32 + c] / n - m * m;
    bnp[c] = (float)m;
    bnp[32 + c] = (float)(1.0 / sqrt(v + 1e-5));
  } else if (c < 56) {
    int cc = c - 32;
    double n = (double)B_SZ * 18;
    double m = dsums[64 + cc] / n;
    double v = dsums[88 + cc] / n - m * m;
    bnp[64 + cc] = (float)m;
    bnp[88 + cc] = (float)(1.0 / sqrt(v + 1e-5));
  }
}

// ---------------- spectral branch: pass2 normalize+relu (recompute conv) ----------------
__global__ void k_spec_apply(const float* __restrict__ ws, const float* __restrict__ w,
                             const float* __restrict__ bias, const float* __restrict__ g,
                             const float* __restrict__ beta, const float* __restrict__ bnp,
                             float* __restrict__ out) {
  int i = blockIdx.x * 256 + threadIdx.x;     // exactly B*NF
  int b = i / NF, l = i - b * NF;
  const float* env = ws + WS_ENV;
  float e[5];
  #pragma unroll
  for (int j = 0; j < 5; ++j) {
    int idx = l + j - 2;
    e[j] = (idx >= 0 && idx < NF) ? env[b * NF + idx] : 0.0f;
  }
  #pragma unroll
  for (int c = 0; c < 32; ++c) {
    float y = bias[c];
    #pragma unroll
    for (int j = 0; j < 5; ++j) y += w[c * 5 + j] * e[j];
    float v = (y - bnp[c]) * bnp[32 + c] * g[c] + beta[c];
    out[OUT_SPECTRAL + ((size_t)b * 32 + c) * NF + l] = fmaxf(v, 0.0f);
  }
}

// ---------------- cardiac finalize: bn+relu+mean, pad 24->32 ----------------
__global__ void k_card_apply(const float* __restrict__ ws, const float* __restrict__ w,
                             const float* __restrict__ bias, const float* __restrict__ g,
                             const float* __restrict__ beta, const float* __restrict__ bnp,
                             float* __restrict__ out) {
  int i = blockIdx.x * 256 + threadIdx.x;     // exactly B*32
  int b = i >> 5, c = i & 31;
  float r = 0.0f;
  if (c < 24) {
    const float* env = ws + WS_ENV + b * NF + 6;
    float acc = 0.0f;
    for (int j = 0; j < 18; ++j) {
      float y = bias[c];
      #pragma unroll
      for (int t = 0; t < 7; ++t) {
        int idx = j + t - 3;
        if (idx >= 0 && idx < 18) y += w[c * 7 + t] * env[idx];
      }
      float v = (y - bnp[64 + c]) * bnp[88 + c] * g[c] + beta[c];
      acc += fmaxf(v, 0.0f);
    }
    r = acc * (1.0f / 18.0f);
  }
  out[OUT_CFEAT + i] = r;
}

// ---------------- resp branch (tiny) ----------------
__global__ void k_resp(const float* __restrict__ ws, const float* __restrict__ w1,
                       const float* __restrict__ b1, const float* __restrict__ w2,
                       const float* __restrict__ b2, float* __restrict__ out) {
  int b = blockIdx.x * 256 + threadIdx.x;     // exactly B
  const float* env = ws + WS_ENV + b * NF;
  float rp[5];
  #pragma unroll
  for (int j = 0; j < 4; ++j) rp[j] = env[j];
  rp[4] = 0.0f;
  float r[16][5];
  #pragma unroll
  for (int c = 0; c < 16; ++c) {
    #pragma unroll
    for (int l = 0; l < 5; ++l) {
      float y = b1[c];
      #pragma unroll
      for (int t = 0; t < 5; ++t) {
        int idx = l + t - 2;
        if (idx >= 0 && idx < 5) y += w1[c * 5 + t] * rp[idx];
      }
      r[c][l] = fmaxf(y, 0.0f);
    }
  }
  for (int co = 0; co < 32; ++co) {
    float acc = 0.0f;
    #pragma unroll
    for (int l = 0; l < 5; ++l) {
      float s = b2[co];
      #pragma unroll
      for (int ci = 0; ci < 16; ++ci) {
        #pragma unroll
        for (int t = 0; t < 3; ++t) {
          int idx = l + t - 1;
          if (idx >= 0 && idx < 5) s += w2[(co * 16 + ci) * 3 + t] * r[ci][idx];
        }
      }
      acc += s;
    }
    out[OUT_RFEAT + b * 32 + co] = acc * 0.2f;
  }
}

extern "C" void kernel_launch(void* const* d_in, const int* in_sizes, int n_in,
                              void* d_out, int out_size, void* d_ws, size_t ws_size,
                              hipStream_t stream) {
  (void)in_sizes; (void)n_in; (void)out_size; (void)ws_size;
  const float* x    = (const float*)d_in[0];
  const float* sw   = (const float*)d_in[1];
  const float* sb   = (const float*)d_in[2];
  const float* sg   = (const float*)d_in[3];
  const float* sbe  = (const float*)d_in[4];
  const float* cw   = (const float*)d_in[5];
  const float* cb   = (const float*)d_in[6];
  const float* cg   = (const float*)d_in[7];
  const float* cbe  = (const float*)d_in[8];
  const float* r1w  = (const float*)d_in[9];
  const float* r1b  = (const float*)d_in[10];
  const float* r2w  = (const float*)d_in[11];
  const float* r2b  = (const float*)d_in[12];
  float* out = (float*)d_out;
  float* ws  = (float*)d_ws;
  double* dsums = (double*)(ws + WS_DSUMF);
  float*  bnp   = ws + WS_BNP;

  // mel center table (numpy-matching double math); w==1 -> filters are row gathers
  MelMap mm;
  {
    double fr = 125.0 / 2.0 / 501.0;
    double pts[63]; int p = 0;
    { double st = (0.5 - 0.1) / 20.0; for (int i = 0; i < 21; ++i) pts[p + i] = 0.1 + st * i; pts[p + 20] = 0.5; p += 21; }
    { double st = (0.8 - 0.5) / 9.0;  for (int i = 0; i < 10; ++i) pts[p + i] = 0.5 + st * i; pts[p + 9] = 0.8;  p += 10; }
    { double st = (3.0 - 0.8) / 31.0; for (int i = 0; i < 32; ++i) pts[p + i] = 0.8 + st * i; pts[p + 31] = 3.0; p += 32; }
    for (int m = 0; m < 63; ++m) mm.c[m] = (int)(pts[m] / fr);
    mm.c[63] = -1;   // pts has only 63 entries -> mel row 63 is all zeros
  }

  k_init <<<1, 128, 0, stream>>>(dsums);
  k_trig <<<(NFP * NPS + 255) / 256, 256, 0, stream>>>(ws);
  k_fmean<<<TOTF, 256, 0, stream>>>(x, ws);
  dim3 g_dft(TOTF / 64, NFP / 32);               // 592 x 16
  k_dft  <<<g_dft, 256, 0, stream>>>(x, ws, out);
  k_env  <<<(B_SZ * NF) / 256, 256, 0, stream>>>(out, ws);
  k_mel  <<<(B_SZ * 64 * NT) / 256, 256, 0, stream>>>(out, mm);
  k_spec_sums<<<(B_SZ * NF) / 256, 256, 0, stream>>>(ws, sw, sb, dsums);
  k_card_sums<<<(B_SZ * 18) / 256, 256, 0, stream>>>(ws, cw, cb, dsums);
  k_bnfin<<<1, 64, 0, stream>>>(dsums, bnp);
  k_spec_apply<<<(B_SZ * NF) / 256, 256, 0, stream>>>(ws, sw, sb, sg, sbe, bnp, out);
  k_card_apply<<<(B_SZ * 32) / 256, 256, 0, stream>>>(ws, cw, cb, cg, cbe, bnp, out);
  k_resp <<<B_SZ / 256, 256, 0, stream>>>(ws, r1w, r1b, r2w, r2b, out);
}